// MultiResChamferLoss_67740224192623
// MI455X (gfx1250) — compile-verified
//
#include <hip/hip_runtime.h>

typedef __attribute__((ext_vector_type(2))) float v2f;
typedef __attribute__((ext_vector_type(8))) float v8f;

#define NPTS 1024
#define NTILE (NPTS / 16)

// One workgroup per batch. 256 threads = 8 waves (wave32).
// Computes  sum_i min_j d[i,j] + sum_j min_i d[i,j]  for its batch,
// where d[i,j] = |b_i - a_j|^2 = |b_i|^2 + |a_j|^2 - 2<b_i,a_j>,
// cross term via V_WMMA_F32_16X16X4_F32 (K=3 padded to 4).
//
// LDS holds each cloud pre-paired in WMMA fragment order:
//   region 0: (x,y) per point   -> K-slice {0,1} (lanes 0-15)
//   region 1: (z,0) per point   -> K-slice {2,3-pad} (lanes 16-31)
// so every A/B fragment is ONE aligned ds_load_b64 straight into the
// even-aligned VGPR pair the WMMA consumes (no register shuffles).
__global__ __launch_bounds__(256) void chamfer_batch_kernel(
    const float* __restrict__ a0,   // [B,3,N]
    const float* __restrict__ b0,   // [B,3,N]
    float* __restrict__ partial) {  // [B]
  __shared__ v2f sAv[2 * NPTS];   // [region][point] -> (K0,K1) pair
  __shared__ v2f sBv[2 * NPTS];
  __shared__ float sAA[NPTS];     // |a_p|^2
  __shared__ float sBB[NPTS];     // |b_p|^2
  __shared__ float sWave[8];

  const int batch = blockIdx.x;
  const int tid = threadIdx.x;
  const int lane = tid & 31;
  const int wave = tid >> 5;
  const size_t base = (size_t)batch * 3 * NPTS;

  // Stage clouds into paired LDS layout and compute norms in one pass
  // (global reads are coalesced: contiguous p per channel row).
  for (int p = tid; p < NPTS; p += 256) {
    float ax = a0[base + p], ay = a0[base + NPTS + p], az = a0[base + 2 * NPTS + p];
    sAv[p] = v2f{ax, ay};
    sAv[NPTS + p] = v2f{az, 0.f};
    sAA[p] = ax * ax + ay * ay + az * az;
    float bx = b0[base + p], by = b0[base + NPTS + p], bz = b0[base + 2 * NPTS + p];
    sBv[p] = v2f{bx, by};
    sBv[NPTS + p] = v2f{bz, 0.f};
    sBB[p] = bx * bx + by * by + bz * bz;
  }
  __syncthreads();

  // ISA A-16x4 layout: lane holds point M = lane&15; VGPR v holds
  // K = v + 2*(lane>>4)  ->  half-wave selects the LDS region.
  const int half = lane >> 4;
  const int l16 = lane & 15;
  const int regOff = half * NPTS + l16;  // v2f-granule offset

  float waveAcc = 0.f;

  for (int dir = 0; dir < 2; ++dir) {
    // dir 0: rows = b points (min over a); dir 1: rows = a points (min over b)
    const v2f* rowsL = ((dir == 0) ? sBv : sAv) + regOff;
    const v2f* colsL = ((dir == 0) ? sAv : sBv) + regOff;
    const float* rNorm = (dir == 0) ? sBB : sAA;
    const float* cNormL = ((dir == 0) ? sAA : sBB) + l16;

    for (int iblk = wave; iblk < NTILE; iblk += 8) {
      const int iBase = iblk * 16;

      const v2f afrag = rowsL[iBase];  // one ds_load_b64

      // Row norms for this lane's 8 D rows: M = v + 8*half (constant over j,
      // re-added AFTER the min-reduction, not in the inner loop).
      float rn[8];
#pragma unroll
      for (int v = 0; v < 8; ++v) rn[v] = rNorm[iBase + half * 8 + v];

      float minD[8];
#pragma unroll
      for (int v = 0; v < 8; ++v) minD[v] = 3.4e38f;

#pragma unroll 4
      for (int jblk = 0; jblk < NTILE; ++jblk) {
        const int jBase = jblk * 16;

        const v2f bfrag = colsL[jBase];  // one ds_load_b64

        v8f c = {};
        // (neg_a, A, neg_b, B, c_mod, C, reuse_a, reuse_b)
        c = __builtin_amdgcn_wmma_f32_16x16x4_f32(
            false, afrag, false, bfrag, (short)0, c, false, false);

        const float cn = cNormL[jBase];
#pragma unroll
        for (int v = 0; v < 8; ++v) {
          // d - rn = cn - 2*cross ; rn re-added after reduction.
          minD[v] = fminf(minD[v], fmaf(-2.0f, c[v], cn));
        }
      }

      // Min across the 16 lanes holding the same row (masks stay in-half),
      // then add back the row norm and sum the 16 row-minima.
      float s = 0.f;
#pragma unroll
      for (int v = 0; v < 8; ++v) {
        float m = minD[v];
        m = fminf(m, __shfl_xor(m, 1, 32));
        m = fminf(m, __shfl_xor(m, 2, 32));
        m = fminf(m, __shfl_xor(m, 4, 32));
        m = fminf(m, __shfl_xor(m, 8, 32));
        s += m + rn[v];
      }
      s += __shfl_xor(s, 16, 32);  // rows 0-7 (half0) + rows 8-15 (half1)
      waveAcc += s;                // identical across all 32 lanes
    }
  }

  if (lane == 0) sWave[wave] = waveAcc;
  __syncthreads();
  if (tid == 0) {
    float t = 0.f;
    for (int w = 0; w < 8; ++w) t += sWave[w];  // deterministic order
    partial[batch] = t;
  }
}

__global__ void chamfer_finalize(const float* __restrict__ partial,
                                 float* __restrict__ out, int nbatch,
                                 float scale) {
  if (threadIdx.x == 0 && blockIdx.x == 0) {
    float t = 0.f;
    for (int i = 0; i < nbatch; ++i) t += partial[i];  // deterministic order
    out[0] = t * scale;
  }
}

extern "C" void kernel_launch(void* const* d_in, const int* in_sizes, int n_in,
                              void* d_out, int out_size, void* d_ws,
                              size_t ws_size, hipStream_t stream) {
  const float* a0 = (const float*)d_in[0];
  const float* b0 = (const float*)d_in[1];
  float* out = (float*)d_out;
  float* partial = (float*)d_ws;

  const int B = in_sizes[0] / (3 * NPTS);  // 64 for the reference shapes

  chamfer_batch_kernel<<<B, 256, 0, stream>>>(a0, b0, partial);
  const float scale = 1000.0f / ((float)NPTS * (float)B);
  chamfer_finalize<<<1, 32, 0, stream>>>(partial, out, B, scale);
}